// PIELMPolyModel_58866821759086
// MI455X (gfx1250) — compile-verified
//
#include <hip/hip_runtime.h>
#include <hip/hip_bf16.h>

// ---------------------------------------------------------------------------
// PIELM poly model on MI455X (gfx1250, wave32).
// Strategy: normal equations G = H^T H (56x56, padded 64x64 with b appended as
// column 56) accumulated with V_WMMA_F32_16X16X4_F32; MLP hidden layer also on
// f32 WMMA; tiny Gauss-Jordan solve; then u_pred = phi @ c.
// ---------------------------------------------------------------------------

typedef __attribute__((ext_vector_type(2))) float v2f;
typedef __attribute__((ext_vector_type(8))) float v8f;

#define TILE_LD 68  // bank-skewed row stride for 32x64 LDS tiles (64 banks)

__device__ __forceinline__ v8f wmma4(v2f a, v2f b, v8f c) {
  // D(16x16 f32) = A(16x4 f32) x B(4x16 f32) + C
  return __builtin_amdgcn_wmma_f32_16x16x4_f32(false, a, false, b, (short)0, c,
                                               false, false);
}

__device__ __forceinline__ void pows5(float x, float* t) {
  t[0] = 1.0f;
#pragma unroll
  for (int e = 1; e <= 5; ++e) t[e] = t[e - 1] * x;
}

// Stage one row of H (64 floats incl. padding cols handled by caller) into LDS.
// WITHLAP: v = muM * lap(phi_m) + rw2M * phi_m   (A_pde row; muM/rw2M pre-masked)
// else   : v = scale * phi_m                      (phi row; scale pre-masked)
template <bool WITHLAP>
__device__ __forceinline__ void stage_row(float* row, const float* px,
                                          const float* py, const float* pz,
                                          float muM, float rw2M, float scale) {
  int m = 0;
#pragma unroll
  for (int t = 0; t <= 5; ++t) {
#pragma unroll
    for (int i = 0; i <= t; ++i) {
#pragma unroll
      for (int j = 0; i + j <= t; ++j) {
        const int k = t - i - j;
        const float phi = px[i] * py[j] * pz[k];
        float v;
        if (WITHLAP) {
          float lap = 0.0f;
          if (i >= 2) lap += (float)(i * (i - 1)) * px[i - 2] * py[j] * pz[k];
          if (j >= 2) lap += (float)(j * (j - 1)) * px[i] * py[j - 2] * pz[k];
          if (k >= 2) lap += (float)(k * (k - 1)) * px[i] * py[j] * pz[k - 2];
          v = muM * lap + rw2M * phi;
        } else {
          v = scale * phi;
        }
        row[m++] = v;
      }
    }
  }
}

__device__ __forceinline__ float eval_dot(const float* px, const float* py,
                                          const float* pz, const float* sc) {
  int m = 0;
  float u = 0.0f;
#pragma unroll
  for (int t = 0; t <= 5; ++t) {
#pragma unroll
    for (int i = 0; i <= t; ++i) {
#pragma unroll
      for (int j = 0; i + j <= t; ++j) {
        const int k = t - i - j;
        u = fmaf(px[i] * py[j] * pz[k], sc[m], u);
        ++m;
      }
    }
  }
  return u;
}

// Build one WMMA operand (f32 16x4 A-layout / 4x16 B-layout, symmetric):
// lane%16 -> matrix row/col within tile, half-wave + vgpr slot -> K index.
__device__ __forceinline__ v2f ld_op(const float* tile, int krow, int c) {
  v2f o;
  o.x = tile[krow * TILE_LD + c];
  o.y = tile[(krow + 1) * TILE_LD + c];
  return o;
}

// Accumulate G += chunk^T * chunk for a 32x64 row chunk staged in LDS.
// Only the 10 upper-triangle 16x16 tiles of the 64x64 result are kept.
__device__ __forceinline__ void gram_accum(const float* tile, v8f* acc, int hf,
                                           int col) {
#pragma unroll
  for (int ks = 0; ks < 8; ++ks) {  // 32 rows, K=4 per WMMA
    const int k0 = ks * 4 + 2 * hf;
    v2f op[4];
#pragma unroll
    for (int t = 0; t < 4; ++t) op[t] = ld_op(tile, k0, t * 16 + col);
    int p = 0;
#pragma unroll
    for (int ti = 0; ti < 4; ++ti) {
#pragma unroll
      for (int tj = ti; tj < 4; ++tj) {
        acc[p] = wmma4(op[ti], op[tj], acc[p]);
        ++p;
      }
    }
  }
}

__device__ __forceinline__ void gram_flush(float* G, const v8f* acc, int hf,
                                           int col) {
  int p = 0;
#pragma unroll
  for (int ti = 0; ti < 4; ++ti) {
#pragma unroll
    for (int tj = ti; tj < 4; ++tj) {
#pragma unroll
      for (int e = 0; e < 8; ++e) {
        const int gi = ti * 16 + e + 8 * hf;  // C/D layout: M = vgpr + 8*half
        const int gj = tj * 16 + col;         // N = lane%16
        atomicAdd(&G[gi * 64 + gj], acc[p][e]);
      }
      ++p;
    }
  }
}

// ---------------------------------------------------------------------------
// Kernel 1: zero workspace (G 64x64 + c 64)
// ---------------------------------------------------------------------------
__global__ void init_ws_kernel(float* ws, int n) {
  const int i = blockIdx.x * blockDim.x + threadIdx.x;
  if (i < n) ws[i] = 0.0f;
}

// ---------------------------------------------------------------------------
// Kernel 2: per-point MLP (mu) + A_pde rows + 0.1*phi rows -> Gram accumulate.
// Block = 128 threads = 4 waves; each wave owns a 32x64 LDS tile.
// ---------------------------------------------------------------------------
__global__ void __launch_bounds__(128) pde_data_gram_kernel(
    const float* __restrict__ X, const float* __restrict__ Udata,
    const float* __restrict__ W1, const float* __restrict__ B1,
    const float* __restrict__ W2, const float* __restrict__ B2,
    const float* __restrict__ W3, const float* __restrict__ B3,
    const int* __restrict__ RW2, float* __restrict__ G,
    float* __restrict__ outMu, int N) {
  __shared__ float sTile[4][32 * TILE_LD];
  __shared__ float sW2[64 * TILE_LD];
  __shared__ float sW1[3 * 64];
  __shared__ float sB1[64];
  __shared__ float sB2[64];
  __shared__ float sW3[64];
  __shared__ float sB3s[1];

  const int tid = threadIdx.x;
  const int wave = tid >> 5;
  const int lane = tid & 31;
  const int hf = lane >> 4;
  const int col = lane & 15;
  float* tile = &sTile[wave][0];

  for (int i = tid; i < 64 * 64; i += 128)
    sW2[(i >> 6) * TILE_LD + (i & 63)] = W2[i];
  for (int i = tid; i < 192; i += 128) sW1[i] = W1[i];
  if (tid < 64) {
    sB1[tid] = B1[tid];
    sB2[tid] = B2[tid];
    sW3[tid] = W3[tid];
  }
  if (tid == 0) sB3s[0] = B3[0];
  __syncthreads();

  const float rw2 = (float)RW2[0];

  v8f acc[10];
#pragma unroll
  for (int p = 0; p < 10; ++p)
#pragma unroll
    for (int e = 0; e < 8; ++e) acc[p][e] = 0.0f;

  const int stride = gridDim.x * 128;
  for (int base = blockIdx.x * 128; base < N; base += stride) {
    const int p = base + wave * 32 + lane;
    const bool valid = (p < N);
    float x0 = 0.0f, x1 = 0.0f, x2 = 0.0f;
    if (valid) {
      x0 = X[p * 3 + 0];
      x1 = X[p * 3 + 1];
      x2 = X[p * 3 + 2];
    }

    // ---- MLP layer 1 (per-lane VALU), stage H1 rows (row = lane) ----
#pragma unroll 4
    for (int n = 0; n < 64; ++n) {
      float h = sB1[n];
      h = fmaf(x0, sW1[n], h);
      h = fmaf(x1, sW1[64 + n], h);
      h = fmaf(x2, sW1[128 + n], h);
      tile[lane * TILE_LD + n] = tanhf(h);
    }
    __syncthreads();

    // ---- MLP layer 2: H2 = tanh(H1 @ W2 + b2) via f32 WMMA ----
#pragma unroll
    for (int sub = 0; sub < 2; ++sub) {  // rows [0..15], [16..31]
      v8f hacc[4];
#pragma unroll
      for (int tn = 0; tn < 4; ++tn) {
        const float b2v = sB2[tn * 16 + col];
#pragma unroll
        for (int e = 0; e < 8; ++e) hacc[tn][e] = b2v;
      }
      const int arow = (sub * 16 + col) * TILE_LD;
#pragma unroll 4
      for (int ks = 0; ks < 16; ++ks) {
        const int k0 = ks * 4 + 2 * hf;
        v2f a;
        a.x = tile[arow + k0];
        a.y = tile[arow + k0 + 1];
#pragma unroll
        for (int tn = 0; tn < 4; ++tn) {
          v2f b;
          b.x = sW2[k0 * TILE_LD + tn * 16 + col];
          b.y = sW2[(k0 + 1) * TILE_LD + tn * 16 + col];
          hacc[tn] = wmma4(a, b, hacc[tn]);
        }
      }
      // tanh + writeback in D-layout (rows sub*16 .. sub*16+15; disjoint from
      // the H1 rows still read by the other sub-chunk)
#pragma unroll
      for (int tn = 0; tn < 4; ++tn)
#pragma unroll
        for (int e = 0; e < 8; ++e)
          tile[(sub * 16 + e + 8 * hf) * TILE_LD + tn * 16 + col] =
              tanhf(hacc[tn][e]);
    }
    __syncthreads();

    // ---- layer 3: mu = H2row . W3 + b3 (per-lane, own row) ----
    float mu = sB3s[0];
#pragma unroll 8
    for (int f = 0; f < 64; ++f)
      mu = fmaf(tile[lane * TILE_LD + f], sW3[f], mu);
    if (valid) outMu[p] = mu;

    float px[6], py[6], pz[6];
    pows5(x0, px);
    pows5(x1, py);
    pows5(x2, pz);
    const float vmask = valid ? 1.0f : 0.0f;

    // ---- stage A_pde rows (b column = 0), accumulate Gram ----
    stage_row<true>(tile + lane * TILE_LD, px, py, pz, mu * vmask, rw2 * vmask,
                    0.0f);
#pragma unroll
    for (int j = 56; j < 64; ++j) tile[lane * TILE_LD + j] = 0.0f;
    __syncthreads();
    gram_accum(tile, acc, hf, col);
    __syncthreads();

    // ---- stage 0.1*phi rows with b column = 0.1*u_data, accumulate ----
    const float ud = valid ? Udata[p] : 0.0f;
    stage_row<false>(tile + lane * TILE_LD, px, py, pz, 0.0f, 0.0f,
                     0.1f * vmask);
    tile[lane * TILE_LD + 56] = 0.1f * ud;
#pragma unroll
    for (int j = 57; j < 64; ++j) tile[lane * TILE_LD + j] = 0.0f;
    __syncthreads();
    gram_accum(tile, acc, hf, col);
    __syncthreads();
  }

  gram_flush(G, acc, hf, col);
}

// ---------------------------------------------------------------------------
// Kernel 3: boundary rows 100*phi(x[bc_idx]) with b column = 100*u_bc.
// ---------------------------------------------------------------------------
__global__ void __launch_bounds__(128) bc_gram_kernel(
    const float* __restrict__ X, const float* __restrict__ Ubc,
    const int* __restrict__ BCI, float* __restrict__ G, int K) {
  __shared__ float sTile[4][32 * TILE_LD];
  const int tid = threadIdx.x;
  const int wave = tid >> 5;
  const int lane = tid & 31;
  const int hf = lane >> 4;
  const int col = lane & 15;
  float* tile = &sTile[wave][0];

  v8f acc[10];
#pragma unroll
  for (int p = 0; p < 10; ++p)
#pragma unroll
    for (int e = 0; e < 8; ++e) acc[p][e] = 0.0f;

  const int stride = gridDim.x * 128;
  for (int base = blockIdx.x * 128; base < K; base += stride) {
    const int r = base + wave * 32 + lane;
    const bool valid = (r < K);
    const int idx = valid ? BCI[r] : 0;
    const float x0 = X[idx * 3 + 0];
    const float x1 = X[idx * 3 + 1];
    const float x2 = X[idx * 3 + 2];
    const float ub = valid ? Ubc[r] : 0.0f;
    const float vmask = valid ? 1.0f : 0.0f;

    float px[6], py[6], pz[6];
    pows5(x0, px);
    pows5(x1, py);
    pows5(x2, pz);

    stage_row<false>(tile + lane * TILE_LD, px, py, pz, 0.0f, 0.0f,
                     100.0f * vmask);
    tile[lane * TILE_LD + 56] = 100.0f * ub;
#pragma unroll
    for (int j = 57; j < 64; ++j) tile[lane * TILE_LD + j] = 0.0f;
    __syncthreads();
    gram_accum(tile, acc, hf, col);
    __syncthreads();
  }

  gram_flush(G, acc, hf, col);
}

// ---------------------------------------------------------------------------
// Kernel 4: solve G c = g (56x56, SPD) via Gauss-Jordan in LDS. Only the
// upper-triangle tiles of G were stored; mirror on read. g = column 56.
// ---------------------------------------------------------------------------
__global__ void __launch_bounds__(64) solve_kernel(const float* __restrict__ G,
                                                   float* __restrict__ C) {
  __shared__ float A[56][58];
  const int r = threadIdx.x;
  if (r < 56) {
    for (int j = 0; j < 56; ++j)
      A[r][j] = ((r >> 4) <= (j >> 4)) ? G[r * 64 + j] : G[j * 64 + r];
    A[r][56] = G[r * 64 + 56];
    A[r][r] *= 1.000001f;  // tiny ridge for conditioning
  }
  __syncthreads();
  for (int k = 0; k < 56; ++k) {
    if (r < 56 && r != k) {
      const float f = A[r][k] / A[k][k];
      for (int j = k; j <= 56; ++j) A[r][j] -= f * A[k][j];
    }
    __syncthreads();
  }
  if (r < 56)
    C[r] = A[r][56] / A[r][r];
  else if (r < 64)
    C[r] = 0.0f;
}

// ---------------------------------------------------------------------------
// Kernel 5: u_pred = phi(x) . c
// ---------------------------------------------------------------------------
__global__ void __launch_bounds__(256) predict_kernel(
    const float* __restrict__ X, const float* __restrict__ C,
    float* __restrict__ out, int N) {
  __shared__ float sc[56];
  if (threadIdx.x < 56) sc[threadIdx.x] = C[threadIdx.x];
  __syncthreads();
  const int p = blockIdx.x * 256 + threadIdx.x;
  if (p >= N) return;
  float px[6], py[6], pz[6];
  pows5(X[p * 3 + 0], px);
  pows5(X[p * 3 + 1], py);
  pows5(X[p * 3 + 2], pz);
  out[p] = eval_dot(px, py, pz, sc);
}

// ---------------------------------------------------------------------------
extern "C" void kernel_launch(void* const* d_in, const int* in_sizes, int n_in,
                              void* d_out, int out_size, void* d_ws,
                              size_t ws_size, hipStream_t stream) {
  (void)n_in;
  (void)out_size;
  (void)ws_size;
  const float* X = (const float*)d_in[0];
  const float* Ubc = (const float*)d_in[1];
  const float* Ud = (const float*)d_in[2];
  const float* W1 = (const float*)d_in[3];
  const float* B1 = (const float*)d_in[4];
  const float* W2 = (const float*)d_in[5];
  const float* B2 = (const float*)d_in[6];
  const float* W3 = (const float*)d_in[7];
  const float* B3 = (const float*)d_in[8];
  const int* BCI = (const int*)d_in[9];
  const int* RW2 = (const int*)d_in[10];

  const int N = in_sizes[0] / 3;
  const int K = in_sizes[9];

  float* out = (float*)d_out;  // [0..N) = u_pred, [N..2N) = mu
  float* G = (float*)d_ws;     // 64x64 padded normal matrix (+ g in col 56)
  float* C = G + 64 * 64;      // 64 coefficients

  init_ws_kernel<<<(64 * 64 + 64 + 255) / 256, 256, 0, stream>>>(G,
                                                                 64 * 64 + 64);
  pde_data_gram_kernel<<<128, 128, 0, stream>>>(X, Ud, W1, B1, W2, B2, W3, B3,
                                                RW2, G, out + N, N);
  bc_gram_kernel<<<16, 128, 0, stream>>>(X, Ubc, BCI, G, K);
  solve_kernel<<<1, 64, 0, stream>>>(G, C);
  predict_kernel<<<(N + 255) / 256, 256, 0, stream>>>(X, C, out, N);
}